// LBNLayer_75436805587268
// MI455X (gfx1250) — compile-verified
//
#include <hip/hip_runtime.h>
#include <math.h>

typedef __attribute__((ext_vector_type(2))) float v2f;
typedef __attribute__((ext_vector_type(8))) float v8f;

#define NP    16
#define NPAIR 120          // 16*15/2
#define FEAT  424          // 4*16 + 3*120

__global__ __launch_bounds__(256) void
lbn_wmma_kernel(const float* __restrict__ E, const float* __restrict__ px,
                const float* __restrict__ py, const float* __restrict__ pz,
                float* __restrict__ out, int batch)
{
    // per-wave staging: 3 cross matrices of 16x16 f32
    __shared__ float smem[8][3 * 256];

    const int lane   = threadIdx.x & 31;
    const int wave   = threadIdx.x >> 5;
    const int waveId = blockIdx.x * (blockDim.x >> 5) + wave;
    const int nWaves = gridDim.x * (blockDim.x >> 5);

    const bool lo  = lane < NP;        // lanes 0..15 carry the event's 16 particles
    const int  m   = lane & 15;        // particle id / N coordinate in D
    const int  mOf = (lane >> 4) << 3; // D row offset: M = r + (0|8)

    // triu inverse map q -> (i,j), constant per lane across all events.
    // Row i starts at base_i = i*(31-i)/2; 961-8*base_i = (31-2i)^2 exactly.
    int gi[4], gj[4];
#pragma unroll
    for (int g = 0; g < 4; ++g) {
        int q = lane + 32 * g;
        if (q < NPAIR) {
            float s = sqrtf((float)(961 - 8 * q));
            int i = (int)((31.0f - s) * 0.5f);
            int j = q - (i * (31 - i)) / 2 + i + 1;
            gi[g] = i; gj[g] = j;
        } else {
            gi[g] = 0; gj[g] = 0;
        }
    }

    for (int b = waveId; b < batch; b += nWaves) {
        const int ib = b * NP;
        float e = 0.f, x = 0.f, y = 0.f, z = 0.f;
        if (lo) {                       // lanes 16..31 keep zeros -> zero K=2,3 cols
            e = E [ib + m];
            x = px[ib + m];
            y = py[ib + m];
            z = pz[ib + m];
        }

        const int ob = b * FEAT;
        // single features, streamed (NT) stores
        if (lo) {
            __builtin_nontemporal_store(e, &out[ob      + m]);
            __builtin_nontemporal_store(x, &out[ob + 16 + m]);
            __builtin_nontemporal_store(y, &out[ob + 32 + m]);
            __builtin_nontemporal_store(z, &out[ob + 48 + m]);
        }

        // D = A(16x4) x B(4x16):  A cols = [u, -v, 0, 0], B rows = [v, u, 0, 0]
        //  => D[M][N] = u[M]*v[N] - v[M]*u[N]  (antisymmetric pair feature)
        const v8f czero = {};
        v2f a, bb;
        a[0] = x;  a[1] = -y;  bb[0] = y;  bb[1] = x;   // cross_z(px,py)
        v8f dz = __builtin_amdgcn_wmma_f32_16x16x4_f32(
                     false, a, false, bb, (short)0, czero, false, false);
        a[0] = y;  a[1] = -z;  bb[0] = z;  bb[1] = y;   // cross_x(py,pz)
        v8f dx = __builtin_amdgcn_wmma_f32_16x16x4_f32(
                     false, a, false, bb, (short)0, czero, false, false);
        a[0] = z;  a[1] = -x;  bb[0] = x;  bb[1] = z;   // cross_y(pz,px)
        v8f dy = __builtin_amdgcn_wmma_f32_16x16x4_f32(
                     false, a, false, bb, (short)0, czero, false, false);

        // stage full 16x16 D matrices into LDS (lane->col, VGPR->row)
        float* sm = smem[wave];
#pragma unroll
        for (int r = 0; r < 8; ++r) {
            int off = (r + mOf) * 16 + m;
            sm[off]       = dz[r];
            sm[256 + off] = dx[r];
            sm[512 + off] = dy[r];
        }

        // gather upper triangle in triu order -> coalesced 128B NT stores
#pragma unroll
        for (int g = 0; g < 4; ++g) {
            int q = lane + 32 * g;
            if (q < NPAIR) {
                int off = gi[g] * 16 + gj[g];
                __builtin_nontemporal_store(sm[off],
                                            &out[ob + 64 + q]);
                __builtin_nontemporal_store(sm[256 + off],
                                            &out[ob + 64 + NPAIR + q]);
                __builtin_nontemporal_store(sm[512 + off],
                                            &out[ob + 64 + 2 * NPAIR + q]);
            }
        }
    }
}

extern "C" void kernel_launch(void* const* d_in, const int* in_sizes, int n_in,
                              void* d_out, int out_size, void* d_ws, size_t ws_size,
                              hipStream_t stream) {
    const float* E  = (const float*)d_in[0];
    const float* px = (const float*)d_in[1];
    const float* py = (const float*)d_in[2];
    const float* pz = (const float*)d_in[3];
    float* out = (float*)d_out;

    const int batch = in_sizes[0] / NP;     // 262144
    const int threads = 256;                // 8 waves/block (wave32)
    int blocks = 2048;                      // 16384 waves -> 16 events/wave
    const int wavesPerBlock = threads / 32;
    if (blocks * wavesPerBlock > batch)
        blocks = (batch + wavesPerBlock - 1) / wavesPerBlock;

    lbn_wmma_kernel<<<blocks, threads, 0, stream>>>(E, px, py, pz, out, batch);
}